// GEM_90331752169540
// MI455X (gfx1250) — compile-verified
//
#include <hip/hip_runtime.h>

typedef __attribute__((ext_vector_type(2))) float v2f;
typedef __attribute__((ext_vector_type(8))) float v8f;
typedef __attribute__((ext_vector_type(4))) float f4;

#define C_DIM 256          // channels (K and N of both GEMMs)
#define M_DIM 1024         // B*N rows
#define NREP  128          // broadcast factor
#define BN_EPS 1e-5f

// ---------------------------------------------------------------------------
// D[M x 256] = A[M x 256] * W^T  + bias      (W stored row-major [256 x 256])
// One wave32 per 16x16 output tile, K-loop of 64 x V_WMMA_F32_16X16X4_F32.
// A-layout per ISA: lanes 0-15 -> M=lane, K={0,1}; lanes 16-31 -> K={2,3}.
// B-layout mirrors it with N in place of M, so both operands are a single
// contiguous float2 load per lane per K-step.
// ---------------------------------------------------------------------------
__global__ void __launch_bounds__(256)
gemm_bias_wmma(const float* __restrict__ A,
               const float* __restrict__ W,
               const float* __restrict__ bias,
               float* __restrict__ D)
{
    const int lane  = threadIdx.x & 31;
    const int wave  = threadIdx.x >> 5;
    const int gwave = blockIdx.x * 8 + wave;   // 128 blocks * 8 waves = 1024 tiles
    const int mTile = gwave >> 4;              // 64 row tiles
    const int nTile = gwave & 15;              // 16 col tiles
    const int rowBase = mTile * 16;
    const int colBase = nTile * 16;
    const int half = lane >> 4;                // selects K pair {0,1} vs {2,3}
    const int l15  = lane & 15;

    const float* aPtr = A + (size_t)(rowBase + l15) * C_DIM + 2 * half;
    const float* wPtr = W + (size_t)(colBase + l15) * C_DIM + 2 * half;

    v8f acc = {};
#pragma unroll 8
    for (int k = 0; k < C_DIM; k += 4) {
        v2f a = *(const v2f*)(aPtr + k);
        v2f b = *(const v2f*)(wPtr + k);
        // 8 args: (neg_a, A, neg_b, B, c_mod, C, reuse_a, reuse_b)
        acc = __builtin_amdgcn_wmma_f32_16x16x4_f32(
                  false, a, false, b, (short)0, acc, false, false);
    }

    // C/D layout: VGPR r, lanes 0-15 -> M = r, lanes 16-31 -> M = r + 8; N = lane&15
    const float bcol = bias[colBase + l15];
#pragma unroll
    for (int r = 0; r < 8; ++r) {
        const int row = rowBase + r + 8 * half;
        D[(size_t)row * C_DIM + colBase + l15] = acc[r] + bcol;
    }
}

// ---------------------------------------------------------------------------
// Per-channel batch stats over the 1024 rows of e, folded with gamma/beta into
// scale = gamma * rsqrt(var+eps), shift = beta - mean*scale.
// One block, one thread per channel; loads are coalesced across channels.
// ---------------------------------------------------------------------------
__global__ void __launch_bounds__(C_DIM)
bn_stats(const float* __restrict__ E,
         const float* __restrict__ gamma,
         const float* __restrict__ beta,
         float* __restrict__ scale,
         float* __restrict__ shift)
{
    const int c = threadIdx.x;
    float s = 0.f, sq = 0.f;
    for (int r = 0; r < M_DIM; ++r) {
        float e = E[(size_t)r * C_DIM + c];
        s  += e;
        sq += e * e;
    }
    const float mean = s * (1.0f / M_DIM);
    const float var  = sq * (1.0f / M_DIM) - mean * mean;   // biased var
    const float sc   = gamma[c] * rsqrtf(var + BN_EPS);
    scale[c] = sc;
    shift[c] = beta[c] - mean * sc;
}

// ---------------------------------------------------------------------------
// out[b, i*N + j, c] = ehat[b, i, c].  One block per source row (b,i): stage
// the normalized 256-float row in LDS once, then stream 128 contiguous copies
// as float4 (GLOBAL_STORE_B128) — this kernel is the 128 MiB HBM-write
// bottleneck, everything else is noise.
// ---------------------------------------------------------------------------
__global__ void __launch_bounds__(C_DIM)
bn_broadcast(const float* __restrict__ E,
             const float* __restrict__ scale,
             const float* __restrict__ shift,
             float* __restrict__ out)
{
    __shared__ float row[C_DIM];
    const int bi  = blockIdx.x;           // 0 .. 1023  (= b*128 + i)
    const int tid = threadIdx.x;

    row[tid] = E[(size_t)bi * C_DIM + tid] * scale[tid] + shift[tid];
    __syncthreads();

    const f4* r4 = (const f4*)row;                      // 64 float4 per row
    f4* o = (f4*)(out + (size_t)bi * NREP * C_DIM);     // 8192 float4 slots
#pragma unroll 4
    for (int t = tid; t < NREP * (C_DIM / 4); t += C_DIM) {
        o[t] = r4[t & 63];                              // j = t/64, q = t&63
    }
}

// ---------------------------------------------------------------------------
extern "C" void kernel_launch(void* const* d_in, const int* in_sizes, int n_in,
                              void* d_out, int out_size, void* d_ws, size_t ws_size,
                              hipStream_t stream)
{
    const float* x     = (const float*)d_in[0];
    const float* Wv    = (const float*)d_in[1];
    const float* bv    = (const float*)d_in[2];
    const float* We    = (const float*)d_in[3];
    const float* be    = (const float*)d_in[4];
    const float* gamma = (const float*)d_in[5];
    const float* beta  = (const float*)d_in[6];
    float* out = (float*)d_out;

    float* v     = (float*)d_ws;                 // 1024*256 f32 = 1 MiB
    float* e     = v + (size_t)M_DIM * C_DIM;    // 1 MiB
    float* scale = e + (size_t)M_DIM * C_DIM;    // 1 KiB
    float* shift = scale + C_DIM;                // 1 KiB

    gemm_bias_wmma<<<128, 256, 0, stream>>>(x, Wv, bv, v);
    gemm_bias_wmma<<<128, 256, 0, stream>>>(v, We, be, e);
    bn_stats     <<<1,   C_DIM, 0, stream>>>(e, gamma, beta, scale, shift);
    bn_broadcast <<<M_DIM, C_DIM, 0, stream>>>(e, scale, shift, out);
}